// Upscale_18811956756989
// MI455X (gfx1250) — compile-verified
//
#include <hip/hip_runtime.h>

// upfirdn2d(x, outer([1,3,3,1],[1,3,3,1])/64*4, up=2, down=1, pad=(2,1))
// Polyphase form: per-axis taps [0.25, 0.75, 0.75, 0.25]
//   out[2m]   = 0.25*x[m-1] + 0.75*x[m]
//   out[2m+1] = 0.75*x[m]   + 0.25*x[m+1]
// applied separably in H and W; out-of-range inputs are zero.
//
// data: [8*64, 256, 256] f32  ->  out: [8*64, 512, 512] f32
// Memory-bound (640 MB traffic, ~4 FMA/output). CDNA5 paths used:
//   - global_load_async_to_lds_b128 staging (+ s_wait_asynccnt)
//   - non-temporal (TH=NT) output stores so the 512 MB write stream does not
//     evict the 128 MB input (which fits in the 192 MB L2 and is reused by
//     neighboring row-tiles as halo).

#define IN_H   256
#define IN_W   256
#define TILE_H 8
#define NIMG   (8 * 64)

typedef int v4i __attribute__((vector_size(4 * sizeof(int))));
typedef float v2f __attribute__((ext_vector_type(2)));
typedef __attribute__((address_space(1))) v4i* gl_v4i_p;   // global
typedef __attribute__((address_space(3))) v4i* ld_v4i_p;   // LDS

// ---- gfx1250 async global->LDS copy of 16 bytes (per active lane) ----
__device__ __forceinline__ void async_copy16(const float* g, float* l) {
#if __has_builtin(__builtin_amdgcn_global_load_async_to_lds_b128)
  __builtin_amdgcn_global_load_async_to_lds_b128(
      (gl_v4i_p)g, (ld_v4i_p)l, /*offset=*/0, /*cpol=*/0);
#else
  asm volatile("global_load_async_to_lds_b128 %0, %1, off"
               :
               : "v"((ld_v4i_p)l), "v"(g)
               : "memory");
#endif
}

__device__ __forceinline__ void wait_async0() {
#if __has_builtin(__builtin_amdgcn_s_wait_asynccnt)
  __builtin_amdgcn_s_wait_asynccnt(0);
#else
  asm volatile("s_wait_asynccnt 0" ::: "memory");
#endif
}

__global__ __launch_bounds__(256) void upfirdn_up2_tile_kernel(
    const float* __restrict__ in, float* __restrict__ out) {
  // tile rows lr = 0..9 hold global rows r0-1 .. r0+8 (zero outside image)
  __shared__ __align__(16) float tile[TILE_H + 2][IN_W];

  const int tid = threadIdx.x;
  const int n   = blockIdx.y;            // image index (B*C)
  const int r0  = blockIdx.x * TILE_H;   // first input row of this tile

  const float* img  = in  + (size_t)n * (IN_H * IN_W);
  float*       oimg = out + (size_t)n * (4 * IN_H * IN_W);  // 512*512

  // ---- stage (TILE_H+2) rows into LDS with async b128 copies ----
  // 10 rows * 64 sixteen-byte segments = 640 copies over 256 threads
  for (int i = tid; i < (TILE_H + 2) * (IN_W / 4); i += 256) {
    const int lr  = i >> 6;       // IN_W/4 == 64 segments per row
    const int seg = i & 63;
    const int gr  = r0 - 1 + lr;
    float* ldst = &tile[lr][seg * 4];
    if (0 <= gr && gr < IN_H) {
      async_copy16(img + (size_t)gr * IN_W + (size_t)seg * 4, ldst);
    } else {
      *(float4*)ldst = make_float4(0.f, 0.f, 0.f, 0.f);  // zero halo row
    }
  }
  wait_async0();
  __syncthreads();

  // ---- polyphase 2x upsample; thread tid owns input column c ----
  const int  c     = tid;
  const bool has_l = (c > 0);
  const bool has_r = (c < IN_W - 1);

  auto hfilt = [&](int lr, float& he, float& ho) {
    const float xm = has_l ? tile[lr][c - 1] : 0.f;
    const float x0 = tile[lr][c];
    const float xp = has_r ? tile[lr][c + 1] : 0.f;
    he = 0.25f * xm + 0.75f * x0;   // output column 2c
    ho = 0.75f * x0 + 0.25f * xp;   // output column 2c+1
  };

  float heP, hoP, heC, hoC, heN, hoN;   // rows r-1, r, r+1 (rolling)
  hfilt(0, heP, hoP);
  hfilt(1, heC, hoC);

  for (int j = 0; j < TILE_H; ++j) {
    hfilt(j + 2, heN, hoN);
    const int r = r0 + j;

    v2f ev, od;                         // output rows 2r and 2r+1
    ev.x = 0.25f * heP + 0.75f * heC;
    ev.y = 0.25f * hoP + 0.75f * hoC;
    od.x = 0.75f * heC + 0.25f * heN;
    od.y = 0.75f * hoC + 0.25f * hoN;

    v2f* orow0 = (v2f*)(oimg + (size_t)(2 * r)     * (2 * IN_W));
    v2f* orow1 = (v2f*)(oimg + (size_t)(2 * r + 1) * (2 * IN_W));
    __builtin_nontemporal_store(ev, orow0 + c);  // global_store_b64 TH=NT
    __builtin_nontemporal_store(od, orow1 + c);

    heP = heC; hoP = hoC;
    heC = heN; hoC = hoN;
  }
}

extern "C" void kernel_launch(void* const* d_in, const int* in_sizes, int n_in,
                              void* d_out, int out_size, void* d_ws, size_t ws_size,
                              hipStream_t stream) {
  const float* data = (const float*)d_in[0];
  // d_in[1] is the 4x4 FIR kernel; its values are compile-time constants here
  // (exactly outer([1,3,3,1],[1,3,3,1])/64 * 4, separable taps [.25,.75,.75,.25]).
  float* out = (float*)d_out;

  dim3 grid(IN_H / TILE_H, NIMG);   // 32 x 512 blocks
  upfirdn_up2_tile_kernel<<<grid, 256, 0, stream>>>(data, out);
}